// Model_71751723647377
// MI455X (gfx1250) — compile-verified
//
#include <hip/hip_runtime.h>

typedef float v2f __attribute__((ext_vector_type(2)));
typedef float v8f __attribute__((ext_vector_type(8)));

#define N_NODES 100000
#define N_EDGES 1600000
#define IN_FEAT 128
#define HIDDEN  64

// ---- workspace layout (in floats) ----
#define DEG_OFF   0
#define MSG1_OFF  102400                       // deg (100000) rounded up
#define MSG1_SIZE (N_NODES * IN_FEAT)          // 12,800,000
#define H_OFF     (MSG1_OFF + MSG1_SIZE)
#define H_SIZE    (N_NODES * HIDDEN)           // 6,400,000
#define MSG2_OFF  (H_OFF + H_SIZE)
#define MSG2_SIZE (N_NODES * HIDDEN)
#define WS_FLOATS (MSG2_OFF + MSG2_SIZE)       // 25,702,400 floats (~103 MB)

// ------------------------------------------------------------------
// Zero the workspace region (deg + msg buffers; h is fully overwritten)
// ------------------------------------------------------------------
__global__ void zero_ws_kernel(float4* __restrict__ p, int n4) {
    int i = blockIdx.x * blockDim.x + threadIdx.x;
    if (i < n4) p[i] = make_float4(0.f, 0.f, 0.f, 0.f);
}

// ------------------------------------------------------------------
// deg[dst] += 1 for every edge
// ------------------------------------------------------------------
__global__ void degree_kernel(const long long* __restrict__ dst,
                              float* __restrict__ deg, int n_edges) {
    int e = blockIdx.x * blockDim.x + threadIdx.x;
    if (e < n_edges) atomicAdd(&deg[dst[e]], 1.0f);
}

// ------------------------------------------------------------------
// msg[dst, :] += feat[src, :]   (one wave32 per edge, lanes stride features)
// F is 128 (layer1) or 64 (layer2). Gather reads are full rows (streaming-
// friendly); atomic adds land in the L2-resident accumulator.
// ------------------------------------------------------------------
__global__ void scatter_add_kernel(const float* __restrict__ feat,
                                   const long long* __restrict__ src,
                                   const long long* __restrict__ dst,
                                   float* __restrict__ msg,
                                   int n_edges, int F) {
    int t = blockIdx.x * blockDim.x + threadIdx.x;
    int e    = t >> 5;
    int lane = t & 31;
    if (e >= n_edges) return;
    long long s = src[e];
    long long d = dst[e];
    const float* fs = feat + (size_t)s * F;
    float*       md = msg  + (size_t)d * F;
    #pragma unroll 4
    for (int f = lane; f < F; f += 32) {
        atomicAdd(&md[f], fs[f]);
    }
}

// ------------------------------------------------------------------
// out = [ msg/deg | x ] @ [ Wl ; Wr ].T + bias   (optional ReLU)
//
// One wave computes a 16x16 output tile via V_WMMA_F32_16X16X4_F32,
// accumulating over K2 = 2*K. Weights staged transposed in LDS
// (stride 65 to avoid bank conflicts). Block: 512 threads = 16 waves
// = 4 row-tiles x 4 col-tiles = 64 nodes.
//
// f32 WMMA A-layout (16x4):  lane<16 holds row=lane, K={0,1};
//                            lane>=16 holds row=lane-16, K={2,3}.
// C/D layout: VGPR i, lanes 0-15 -> M=i, N=lane; lanes 16-31 -> M=i+8.
// ------------------------------------------------------------------
__global__ void sage_gemm_wmma(const float* __restrict__ msg,
                               const float* __restrict__ deg,
                               const float* __restrict__ xin,
                               const float* __restrict__ Wl,   // [H][K]
                               const float* __restrict__ Wr,   // [H][K]
                               const float* __restrict__ bias, // [H]
                               float* __restrict__ out,        // [N][H]
                               int K, int do_relu) {
    extern __shared__ float Bs[];          // [2K][65]
    const int BSTR = HIDDEN + 1;           // 65 : conflict-free padding
    const int tid  = threadIdx.x;

    // Stage B = [Wl ; Wr] transposed: Bs[k][n] = Wl[n][k], Bs[K+k][n] = Wr[n][k]
    const int total = K * HIDDEN;
    for (int idx = tid; idx < total; idx += blockDim.x) {
        int n = idx / K;
        int k = idx - n * K;
        Bs[k * BSTR + n]       = Wl[idx];
        Bs[(K + k) * BSTR + n] = Wr[idx];
    }
    __syncthreads();

    const int wave     = tid >> 5;
    const int lane     = tid & 31;
    const int row_tile = wave >> 2;        // 0..3
    const int col_tile = wave & 3;         // 0..3
    const long long row_base = (long long)blockIdx.x * 64 + row_tile * 16;
    if (row_base >= N_NODES) return;       // wave-uniform: EXEC stays all-ones

    const int m     = lane & 15;
    const int khalf = (lane >> 4) * 2;     // 0 for lanes 0-15, 2 for 16-31
    const long long row = row_base + m;
    const int col       = col_tile * 16 + m;

    const float invd = 1.0f / fmaxf(deg[row], 1.0f);

    v8f acc = {};

    // ---- first half of K2: mean-aggregated neighbor features ----
    const float* arow = msg + row * K;
    for (int kk = 0; kk < K; kk += 4) {
        float2 av = *(const float2*)(arow + kk + khalf);   // 8B-aligned
        v2f a; a.x = av.x * invd; a.y = av.y * invd;
        v2f b;
        b.x = Bs[(kk + khalf) * BSTR + col];
        b.y = Bs[(kk + khalf + 1) * BSTR + col];
        acc = __builtin_amdgcn_wmma_f32_16x16x4_f32(
            false, a, false, b, (short)0, acc, false, false);
    }

    // ---- second half of K2: self features through Wr ----
    const float* xrow = xin + row * K;
    for (int kk = 0; kk < K; kk += 4) {
        float2 av = *(const float2*)(xrow + kk + khalf);
        v2f a; a.x = av.x; a.y = av.y;
        v2f b;
        b.x = Bs[(K + kk + khalf) * BSTR + col];
        b.y = Bs[(K + kk + khalf + 1) * BSTR + col];
        acc = __builtin_amdgcn_wmma_f32_16x16x4_f32(
            false, a, false, b, (short)0, acc, false, false);
    }

    // ---- epilogue: bias, ReLU, scatter 16x16 tile to out ----
    const float bv = bias[col];
    const int moff = (lane >> 4) * 8;
    #pragma unroll
    for (int i = 0; i < 8; ++i) {
        float v = acc[i] + bv;
        if (do_relu) v = fmaxf(v, 0.0f);
        long long node = row_base + moff + i;
        out[node * HIDDEN + col] = v;
    }
}

// ------------------------------------------------------------------
extern "C" void kernel_launch(void* const* d_in, const int* in_sizes, int n_in,
                              void* d_out, int out_size, void* d_ws, size_t ws_size,
                              hipStream_t stream) {
    const float*     x   = (const float*)d_in[0];
    const long long* ei  = (const long long*)d_in[1];   // int64 [2][N_EDGES]
    const float*     Wl1 = (const float*)d_in[2];
    const float*     bl1 = (const float*)d_in[3];
    const float*     Wr1 = (const float*)d_in[4];
    const float*     Wl2 = (const float*)d_in[5];
    const float*     bl2 = (const float*)d_in[6];
    const float*     Wr2 = (const float*)d_in[7];

    float* ws   = (float*)d_ws;
    float* deg  = ws + DEG_OFF;
    float* msg1 = ws + MSG1_OFF;
    float* h    = ws + H_OFF;
    float* msg2 = ws + MSG2_OFF;
    float* outp = (float*)d_out;

    const long long* src = ei;
    const long long* dst = ei + N_EDGES;

    // 1) zero deg + msg buffers (~103 MB, ~4.5us at 23.3 TB/s)
    {
        int n4 = WS_FLOATS / 4;
        int blocks = (n4 + 255) / 256;
        zero_ws_kernel<<<blocks, 256, 0, stream>>>((float4*)ws, n4);
    }

    // 2) degrees
    degree_kernel<<<(N_EDGES + 255) / 256, 256, 0, stream>>>(dst, deg, N_EDGES);

    // 3) layer-1 neighbor sum (128 feats)
    {
        long long threads = (long long)N_EDGES * 32;
        int blocks = (int)((threads + 255) / 256);
        scatter_add_kernel<<<blocks, 256, 0, stream>>>(x, src, dst, msg1,
                                                       N_EDGES, IN_FEAT);
    }

    // 4) h = relu([msg1/deg | x] @ [Wl1;Wr1].T + bl1)
    {
        int blocks = (N_NODES + 63) / 64;
        size_t smem = (size_t)(2 * IN_FEAT) * (HIDDEN + 1) * sizeof(float); // 66560B
        sage_gemm_wmma<<<blocks, 512, smem, stream>>>(msg1, deg, x, Wl1, Wr1,
                                                      bl1, h, IN_FEAT, 1);
    }

    // 5) layer-2 neighbor sum (64 feats)
    {
        long long threads = (long long)N_EDGES * 32;
        int blocks = (int)((threads + 255) / 256);
        scatter_add_kernel<<<blocks, 256, 0, stream>>>(h, src, dst, msg2,
                                                       N_EDGES, HIDDEN);
    }

    // 6) out = [msg2/deg | h] @ [Wl2;Wr2].T + bl2
    {
        int blocks = (N_NODES + 63) / 64;
        size_t smem = (size_t)(2 * HIDDEN) * (HIDDEN + 1) * sizeof(float); // 33280B
        sage_gemm_wmma<<<blocks, 512, smem, stream>>>(msg2, deg, h, Wl2, Wr2,
                                                      bl2, outp, HIDDEN, 0);
    }
}